// GraphUpdateBlock_73572789781100
// MI455X (gfx1250) — compile-verified
//
#include <hip/hip_runtime.h>
#include <hip/hip_bf16.h>

// ---------------------------------------------------------------------------
// GATv2(770->256,H4) -> silu -> GATv2(256->1024,H4) -> proj -> GRU -> 4 heads
// Dense GEMMs: fp32 WMMA (V_WMMA_F32_16X16X4_F32), 16x128 tile per wave with
// 8 accumulators, compile-time K/Nout (immediate-offset loads) and manual
// double-buffering so WMMAs overlap the next iteration's loads.
// Edge softmax: wave-per-edge coalesced gathers + float atomics; working set
// is L2-resident on MI455X (192MB L2).
// ---------------------------------------------------------------------------

#define NNODES 10000
#define NEDGES 160000

typedef float v2f __attribute__((ext_vector_type(2)));
typedef float v8f __attribute__((ext_vector_type(8)));

// ---------------- fill ----------------
__global__ void k_fill(float* __restrict__ p, float v, int n) {
    int i = blockIdx.x * blockDim.x + threadIdx.x;
    if (i < n) p[i] = v;
}

// ---------------- concat x = [c_temp | c_stereo | e_proj | f_Lt | pad] ----------------
__global__ __launch_bounds__(256) void k_concat(
    const float* __restrict__ ct, const float* __restrict__ cs,
    const float* __restrict__ ep, const float* __restrict__ fl,
    float* __restrict__ x) {
    int n = blockIdx.x;
    for (int c = threadIdx.x; c < 784; c += 256) {
        float v;
        if      (c < 256) v = ct[n * 256 + c];
        else if (c < 512) v = cs[n * 256 + (c - 256)];
        else if (c < 514) v = ep[n * 2   + (c - 512)];
        else if (c < 770) v = fl[n * 256 + (c - 514)];
        else              v = 0.0f;
        x[(size_t)n * 784 + c] = v;
    }
}

// ---------------- pad weight rows 770 -> 784 ----------------
__global__ void k_padw(const float* __restrict__ W, float* __restrict__ Wp,
                       int rows, int Kin, int Kout) {
    int i = blockIdx.x * blockDim.x + threadIdx.x;
    if (i >= rows * Kout) return;
    int r = i / Kout, k = i - r * Kout;
    Wp[i] = (k < Kin) ? W[r * Kin + k] : 0.0f;
}

// ---------------- fp32 WMMA GEMM:  C = act(A[M,K] @ W[Nout,K]^T + bias) ----------------
// Wave tile: 16 rows x 128 cols = 8 f32 accumulators. Per K-step of 4:
// 1 A-fragment + 8 B-fragment loads (immediate offsets; K,NOUT constexpr)
// double-buffered against 8 V_WMMA_F32_16X16X4_F32 on the previous fragments.
// A 16x4: lanes 0-15 -> K={k,k+1}, lanes 16-31 -> K={k+2,k+3} (M = lane&15)
// B 4x16: same K split, N = lane&15
// C/D: lanes 0-15 rows 0..7 (vgpr i -> M=i), lanes 16-31 rows 8..15.
template <int K, int NOUT>
__global__ __launch_bounds__(256) void k_gemm_wmma(
    const float* __restrict__ A, const float* __restrict__ W,
    const float* __restrict__ bias, float* __restrict__ C,
    int act, int ntiles) {
    int lane = threadIdx.x & 31;
    int wave = threadIdx.x >> 5;
    int w = blockIdx.x * 8 + wave;
    if (w >= ntiles) return;

    constexpr int NCOL = NOUT >> 7;        // column tiles of 128
    int tm = w / NCOL;
    int tn = w - tm * NCOL;

    int r16   = lane & 15;
    int khalf = (lane >> 4) << 1;          // 0 or 2
    int row = tm * 16 + r16;
    int colbase = tn * 128 + r16;

    const float* arow  = A + (size_t)row * K + khalf;
    const float* wbase = W + (size_t)colbase * K + khalf;   // + j*16*K (immediate)

    v8f acc[8];
#pragma unroll
    for (int j = 0; j < 8; ++j) acc[j] = (v8f){};

    v2f a_cur, b_cur[8];
    a_cur.x = arow[0];  a_cur.y = arow[1];
#pragma unroll
    for (int j = 0; j < 8; ++j) {
        b_cur[j].x = wbase[j * 16 * K];
        b_cur[j].y = wbase[j * 16 * K + 1];
    }

    for (int k = 4; k < K; k += 4) {
        v2f a_nxt, b_nxt[8];
        a_nxt.x = arow[k];  a_nxt.y = arow[k + 1];
#pragma unroll
        for (int j = 0; j < 8; ++j) {
            b_nxt[j].x = wbase[j * 16 * K + k];
            b_nxt[j].y = wbase[j * 16 * K + k + 1];
        }
        __builtin_prefetch(arow + k + 60, 0, 1);   // global_prefetch_b8 (speculative)
#pragma unroll
        for (int j = 0; j < 8; ++j)
            acc[j] = __builtin_amdgcn_wmma_f32_16x16x4_f32(
                false, a_cur, false, b_cur[j], (short)0, acc[j], false, false);
        a_cur = a_nxt;
#pragma unroll
        for (int j = 0; j < 8; ++j) b_cur[j] = b_nxt[j];
    }
#pragma unroll
    for (int j = 0; j < 8; ++j)
        acc[j] = __builtin_amdgcn_wmma_f32_16x16x4_f32(
            false, a_cur, false, b_cur[j], (short)0, acc[j], false, false);

    int rbase = tm * 16 + ((lane >> 4) << 3);   // rows 0..7 or 8..15 of tile
#pragma unroll
    for (int j = 0; j < 8; ++j) {
        int col = colbase + 16 * j;
        float bv = bias[col];
#pragma unroll
        for (int i = 0; i < 8; ++i) {
            float v = acc[j][i] + bv;
            if (act == 1) v = v / (1.0f + expf(-v));     // silu
            C[(size_t)(rbase + i) * NOUT + col] = v;
        }
    }
}

// ---------------- float atomic max (init to -inf) ----------------
__device__ inline void atomicMaxFloat(float* addr, float val) {
    if (val >= 0.0f) atomicMax((int*)addr, __float_as_int(val));
    else             atomicMin((unsigned int*)addr, __float_as_uint(val));
}

// ---------------- edge pass A: wave per edge, coalesced; logits + atomic max ----------
template <int CC>   // channels per head; H = 4
__global__ __launch_bounds__(256) void k_edge_logits(
    const int* __restrict__ edges, const float* __restrict__ ea,
    const float* __restrict__ xl, const float* __restrict__ xr,
    const float* __restrict__ We, const float* __restrict__ att,
    float* __restrict__ logits, float* __restrict__ nmax, int E) {
    int wave = threadIdx.x >> 5;
    int lane = threadIdx.x & 31;
    int e = blockIdx.x * 8 + wave;
    if (e >= E) return;
    int src = edges[e], dst = edges[E + e];
    float a0 = ea[e * 3], a1 = ea[e * 3 + 1], a2 = ea[e * 3 + 2];
    constexpr int HC = 4 * CC;
    const float* xlp = xl + (size_t)src * HC;
    const float* xrp = xr + (size_t)dst * HC;
#pragma unroll
    for (int h = 0; h < 4; ++h) {
        float s = 0.0f;
        for (int cc = lane; cc < CC; cc += 32) {
            int c = h * CC + cc;
            float ev = a0 * We[3 * c] + a1 * We[3 * c + 1] + a2 * We[3 * c + 2];
            float v = xlp[c] + xrp[c] + ev;
            v = (v > 0.0f) ? v : 0.2f * v;                // leaky_relu(0.2)
            s += v * att[c];
        }
        for (int off = 16; off > 0; off >>= 1)
            s += __shfl_down(s, off, 32);
        if (lane == 0) {
            logits[e * 4 + h] = s;
            atomicMaxFloat(&nmax[dst * 4 + h], s);
        }
    }
}

// ---------------- edge pass B: p = exp(logit - max), accumulate denom ----------------
__global__ __launch_bounds__(256) void k_edge_exp(
    const int* __restrict__ edges, float* __restrict__ logits,
    const float* __restrict__ nmax, float* __restrict__ nden, int E) {
    int idx = blockIdx.x * blockDim.x + threadIdx.x;
    if (idx >= E * 4) return;
    int e = idx >> 2, h = idx & 3;
    int dst = edges[E + e];
    float p = expf(logits[idx] - nmax[dst * 4 + h]);
    logits[idx] = p;
    atomicAdd(&nden[dst * 4 + h], p);
}

// ---------------- edge pass C: out[dst] += alpha * xl[src], one block per edge --------
template <int CC>   // channels per head; H = 4
__global__ __launch_bounds__(256) void k_edge_scatter(
    const int* __restrict__ edges, const float* __restrict__ logits,
    const float* __restrict__ nden, const float* __restrict__ xl,
    float* __restrict__ out, int E) {
    int e = blockIdx.x;
    constexpr int HC = 4 * CC;
    constexpr int CSH = (CC == 64) ? 6 : 8;  // log2(CC)
    int src = edges[e], dst = edges[E + e];
#pragma unroll
    for (int c = threadIdx.x; c < HC; c += 256) {
        int h = c >> CSH;
        float alpha = logits[e * 4 + h] / (nden[dst * 4 + h] + 1e-16f);
        atomicAdd(&out[(size_t)dst * HC + c], alpha * xl[(size_t)src * HC + c]);
    }
}

// ---------------- x = act(x + bias) ----------------
__global__ void k_bias_act(float* __restrict__ x, const float* __restrict__ bias,
                           int ncols, int total, int act) {
    int i = blockIdx.x * blockDim.x + threadIdx.x;
    if (i >= total) return;
    float v = x[i] + bias[i % ncols];
    if (act == 1) v = v / (1.0f + expf(-v));          // silu
    x[i] = v;
}

// ---------------- GRU cell ----------------
__global__ void k_gru(const float* __restrict__ gi, const float* __restrict__ gh,
                      const float* __restrict__ h, float* __restrict__ hnew, int total) {
    int i = blockIdx.x * blockDim.x + threadIdx.x;
    if (i >= total) return;
    int n = i >> 8, j = i & 255;
    const float* gin = gi + (size_t)n * 768;
    const float* ghn = gh + (size_t)n * 768;
    float ir = gin[j], iz = gin[256 + j], inn = gin[512 + j];
    float hr = ghn[j], hz = ghn[256 + j], hn_ = ghn[512 + j];
    float rg = 1.0f / (1.0f + expf(-(ir + hr)));
    float zg = 1.0f / (1.0f + expf(-(iz + hz)));
    float ng = tanhf(inn + rg * hn_);
    hnew[i] = (1.0f - zg) * ng + zg * h[i];
}

// ---------------- tiny head layer-2: out[n,j] = act2(t[n,:]·W2[j,:] + b2[j]) ----------
__global__ __launch_bounds__(32) void k_head2(
    const float* __restrict__ t, const float* __restrict__ W2,
    const float* __restrict__ b2, float* __restrict__ out,
    int K, int J, int act) {
    int n = blockIdx.x;
    int lane = threadIdx.x;
    for (int j = 0; j < J; ++j) {
        float s = 0.0f;
        for (int k = lane; k < K; k += 32)
            s += t[(size_t)n * K + k] * W2[j * K + k];
        for (int off = 16; off > 0; off >>= 1)
            s += __shfl_down(s, off, 32);
        if (lane == 0) {
            float v = s + b2[j];
            if      (act == 1) v = 1.0f / (1.0f + expf(-v));          // sigmoid
            else if (act == 2) v = (v > 20.0f) ? v : log1pf(expf(v)); // softplus
            out[(size_t)n * J + j] = v;
        }
    }
}

// ---------------------------------------------------------------------------
extern "C" void kernel_launch(void* const* d_in, const int* in_sizes, int n_in,
                              void* d_out, int out_size, void* d_ws, size_t ws_size,
                              hipStream_t stream) {
    const int N = NNODES, E = NEDGES;

    const float* h       = (const float*)d_in[0];
    const float* c_temp  = (const float*)d_in[1];
    const float* c_ster  = (const float*)d_in[2];
    const float* e_proj  = (const float*)d_in[3];
    const float* f_Lt    = (const float*)d_in[4];
    const int*   edges   = (const int*)  d_in[5];
    const float* e_attr  = (const float*)d_in[6];
    const float* c1_Wl   = (const float*)d_in[7];
    const float* c1_bl   = (const float*)d_in[8];
    const float* c1_Wr   = (const float*)d_in[9];
    const float* c1_br   = (const float*)d_in[10];
    const float* c1_We   = (const float*)d_in[11];
    const float* c1_att  = (const float*)d_in[12];
    const float* c1_bias = (const float*)d_in[13];
    const float* c2_Wl   = (const float*)d_in[14];
    const float* c2_bl   = (const float*)d_in[15];
    const float* c2_Wr   = (const float*)d_in[16];
    const float* c2_br   = (const float*)d_in[17];
    const float* c2_We   = (const float*)d_in[18];
    const float* c2_att  = (const float*)d_in[19];
    const float* c2_bias = (const float*)d_in[20];
    const float* proj_W  = (const float*)d_in[21];
    const float* proj_b  = (const float*)d_in[22];
    const float* gru_Wih = (const float*)d_in[23];
    const float* gru_Whh = (const float*)d_in[24];
    const float* gru_bih = (const float*)d_in[25];
    const float* gru_bhh = (const float*)d_in[26];
    const float* res_W1  = (const float*)d_in[27];
    const float* res_b1  = (const float*)d_in[28];
    const float* res_W2  = (const float*)d_in[29];
    const float* res_b2  = (const float*)d_in[30];
    const float* wgt_W1  = (const float*)d_in[31];
    const float* wgt_b1  = (const float*)d_in[32];
    const float* wgt_W2  = (const float*)d_in[33];
    const float* wgt_b2  = (const float*)d_in[34];
    const float* ap_W1   = (const float*)d_in[35];
    const float* ap_b1   = (const float*)d_in[36];
    const float* ap_W2   = (const float*)d_in[37];
    const float* ap_b2   = (const float*)d_in[38];
    const float* ad_W1   = (const float*)d_in[39];
    const float* ad_b1   = (const float*)d_in[40];
    const float* ad_W2   = (const float*)d_in[41];
    const float* ad_b2   = (const float*)d_in[42];

    float* ws = (float*)d_ws;
    size_t o = 0;
    float* xw    = ws + o; o += (size_t)N * 784;      // concat input (padded)
    float* wpadl = ws + o; o += (size_t)256 * 784;    // padded c1_Wl
    float* wpadr = ws + o; o += (size_t)256 * 784;    // padded c1_Wr
    float* xl1   = ws + o; o += (size_t)N * 256;
    float* xr1   = ws + o; o += (size_t)N * 256;
    float* acc1  = ws + o; o += (size_t)N * 256;      // -> x1
    float* logit = ws + o; o += (size_t)E * 4;
    float* nmax  = ws + o; o += (size_t)N * 4;
    float* nden  = ws + o; o += (size_t)N * 4;
    float* xl2   = ws + o; o += (size_t)N * 1024;     // later reused as gi
    float* xr2   = ws + o; o += (size_t)N * 1024;     // later reused as gh
    float* acc2  = ws + o; o += (size_t)N * 1024;     // -> x2
    float* xs    = ws + o; o += (size_t)N * 256;
    float* thead = ws + o; o += (size_t)N * 256;
    (void)ws_size; (void)in_sizes; (void)n_in; (void)out_size;

    float* out_h  = (float*)d_out;                    // [N,256]
    float* out_r  = out_h + (size_t)N * 256;          // [N,2]
    float* out_w  = out_r + (size_t)N * 2;            // [N,2]
    float* out_ap = out_w + (size_t)N * 2;            // [N,1]
    float* out_ad = out_ap + (size_t)N;               // [N,1]

    const int TB = 256;
    auto cdiv = [](int a, int b) { return (a + b - 1) / b; };
    auto tiles = [&](int Nout) { return (N / 16) * (Nout >> 7); };

    // 1. build padded input + padded conv1 weights
    k_concat<<<N, TB, 0, stream>>>(c_temp, c_ster, e_proj, f_Lt, xw);
    k_padw<<<cdiv(256 * 784, TB), TB, 0, stream>>>(c1_Wl, wpadl, 256, 770, 784);
    k_padw<<<cdiv(256 * 784, TB), TB, 0, stream>>>(c1_Wr, wpadr, 256, 770, 784);

    // 2. conv1 linear layers (WMMA):  [N,784] x [256,784]^T
    int t256 = tiles(256);
    k_gemm_wmma<784, 256><<<cdiv(t256, 8), TB, 0, stream>>>(xw, wpadl, c1_bl, xl1, 0, t256);
    k_gemm_wmma<784, 256><<<cdiv(t256, 8), TB, 0, stream>>>(xw, wpadr, c1_br, xr1, 0, t256);

    // 3. conv1 edge softmax + aggregate
    k_fill<<<cdiv(N * 256, TB), TB, 0, stream>>>(acc1, 0.0f, N * 256);
    k_fill<<<cdiv(N * 4, TB), TB, 0, stream>>>(nmax, -INFINITY, N * 4);
    k_fill<<<cdiv(N * 4, TB), TB, 0, stream>>>(nden, 0.0f, N * 4);
    k_edge_logits<64><<<cdiv(E, 8), TB, 0, stream>>>(edges, e_attr, xl1, xr1,
                                                     c1_We, c1_att, logit, nmax, E);
    k_edge_exp<<<cdiv(E * 4, TB), TB, 0, stream>>>(edges, logit, nmax, nden, E);
    k_edge_scatter<64><<<E, TB, 0, stream>>>(edges, logit, nden, xl1, acc1, E);
    k_bias_act<<<cdiv(N * 256, TB), TB, 0, stream>>>(acc1, c1_bias, 256, N * 256, 1); // silu -> x1

    // 4. conv2 linear layers (WMMA): [N,256] x [1024,256]^T
    int t1024 = tiles(1024);
    k_gemm_wmma<256, 1024><<<cdiv(t1024, 8), TB, 0, stream>>>(acc1, c2_Wl, c2_bl, xl2, 0, t1024);
    k_gemm_wmma<256, 1024><<<cdiv(t1024, 8), TB, 0, stream>>>(acc1, c2_Wr, c2_br, xr2, 0, t1024);

    // 5. conv2 edge softmax + aggregate
    k_fill<<<cdiv(N * 1024, TB), TB, 0, stream>>>(acc2, 0.0f, N * 1024);
    k_fill<<<cdiv(N * 4, TB), TB, 0, stream>>>(nmax, -INFINITY, N * 4);
    k_fill<<<cdiv(N * 4, TB), TB, 0, stream>>>(nden, 0.0f, N * 4);
    k_edge_logits<256><<<cdiv(E, 8), TB, 0, stream>>>(edges, e_attr, xl2, xr2,
                                                      c2_We, c2_att, logit, nmax, E);
    k_edge_exp<<<cdiv(E * 4, TB), TB, 0, stream>>>(edges, logit, nmax, nden, E);
    k_edge_scatter<256><<<E, TB, 0, stream>>>(edges, logit, nden, xl2, acc2, E);
    k_bias_act<<<cdiv(N * 1024, TB), TB, 0, stream>>>(acc2, c2_bias, 1024, N * 1024, 0); // -> x2

    // 6. proj: xs = x2 @ proj_W^T + proj_b
    k_gemm_wmma<1024, 256><<<cdiv(t256, 8), TB, 0, stream>>>(acc2, proj_W, proj_b, xs, 0, t256);

    // 7. GRU gates (reuse xl2/xr2 buffers for gi/gh): Nout = 768
    int t768 = tiles(768);
    k_gemm_wmma<256, 768><<<cdiv(t768, 8), TB, 0, stream>>>(xs, gru_Wih, gru_bih, xl2, 0, t768);
    k_gemm_wmma<256, 768><<<cdiv(t768, 8), TB, 0, stream>>>(h,  gru_Whh, gru_bhh, xr2, 0, t768);
    k_gru<<<cdiv(N * 256, TB), TB, 0, stream>>>(xl2, xr2, h, out_h, N * 256);

    // 8. heads
    int t128 = tiles(128);
    k_gemm_wmma<256, 256><<<cdiv(t256, 8), TB, 0, stream>>>(out_h, res_W1, res_b1, thead, 1, t256);
    k_head2<<<N, 32, 0, stream>>>(thead, res_W2, res_b2, out_r, 256, 2, 0);

    k_gemm_wmma<256, 256><<<cdiv(t256, 8), TB, 0, stream>>>(out_h, wgt_W1, wgt_b1, thead, 1, t256);
    k_head2<<<N, 32, 0, stream>>>(thead, wgt_W2, wgt_b2, out_w, 256, 2, 1);

    k_gemm_wmma<256, 128><<<cdiv(t128, 8), TB, 0, stream>>>(out_h, ap_W1, ap_b1, thead, 1, t128);
    k_head2<<<N, 32, 0, stream>>>(thead, ap_W2, ap_b2, out_ap, 128, 1, 2);

    k_gemm_wmma<256, 128><<<cdiv(t128, 8), TB, 0, stream>>>(out_h, ad_W1, ad_b1, thead, 1, t128);
    k_head2<<<N, 32, 0, stream>>>(thead, ad_W2, ad_b2, out_ad, 128, 1, 2);
}